// DCT_FSR_19267223290301
// MI455X (gfx1250) — compile-verified
//
#include <hip/hip_runtime.h>

// CDNA5 (gfx1250) DCT-based 2x FSR upsample.
// out[b,c] = mask[c] * U @ x[b,c] @ U^T,  U = D128[0:64,:]^T @ D64  (128x64)
// V_WMMA_F32_16X16X4_F32, register-blocked; async global->LDS staging.

typedef __attribute__((ext_vector_type(2))) float v2f;
typedef __attribute__((ext_vector_type(8))) float v8f;
typedef int v4i __attribute__((vector_size(16)));   // matches builtin param type

#define HH   64     // low-res H=W
#define HH2  128    // high-res
#define US   76     // LDS row stride (dwords): 304B rows -> 16B aligned,
                    // conflict-free for both row reads (12*lr+k distinct,
                    // hi-half +2 never aliases) and column reads (2*US=24 mod 64)
#define NCH  256

#if defined(__gfx1250__) && __has_builtin(__builtin_amdgcn_global_load_async_to_lds_b128)
#define USE_ASYNC_LDS 1
#else
#define USE_ASYNC_LDS 0
#endif

#define WMMA_F32(a, b, c) \
    __builtin_amdgcn_wmma_f32_16x16x4_f32(false, (a), false, (b), (short)0, (c), false, false)

// ---------------------------------------------------------------------------
// Build U[n][h] = sum_{k<64} D128[k][n] * D64[k][h]   (row-major 128x64)
// Exact integer angle reduction keeps cosf args in [0, 2*pi).
// ---------------------------------------------------------------------------
__global__ __launch_bounds__(256)
void build_U_kernel(float* __restrict__ U) {
    int idx = blockIdx.x * blockDim.x + threadIdx.x;
    if (idx >= HH2 * HH) return;
    int n = idx >> 6;   // 0..127
    int h = idx & 63;   // 0..63
    const float PI = 3.14159265358979323846f;
    const float s128_0 = 0.08838834764831845f;                        // sqrt(1/128)
    const float s128   = 0.08838834764831845f * 1.41421356237309515f; // sqrt(2/128)
    const float s64_0  = 0.125f;                                      // sqrt(1/64)
    const float s64    = 0.17677669529663687f;                        // sqrt(2/64)
    float acc = 0.0f;
    for (int k = 0; k < 64; ++k) {
        int r1 = ((2 * n + 1) * k) & 511;   // theta1 = pi*r1/256
        int r2 = ((2 * h + 1) * k) & 255;   // theta2 = pi*r2/128
        float a = (k == 0 ? s128_0 : s128) * cosf(PI * (1.0f / 256.0f) * (float)r1);
        float b = (k == 0 ? s64_0  : s64 ) * cosf(PI * (1.0f / 128.0f) * (float)r2);
        acc += a * b;
    }
    U[idx] = acc;
}

// ---------------------------------------------------------------------------
// One workgroup per (b,c) image. 256 threads = 8 waves; wave w owns output
// rows [16w,16w+16). GEMM1: T = U @ x (k0-outer, 4 accumulators).
// GEMM2: out = T @ U^T (k0-outer, 8 accumulators). Wave w's GEMM2 A-strip and
// K-span are exactly its own GEMM1 tiles -> no barrier needed between GEMMs.
// ---------------------------------------------------------------------------
__global__ __launch_bounds__(256)
void dct_fsr_kernel(const float* __restrict__ x,
                    const float* __restrict__ mask,
                    const float* __restrict__ Ug,
                    float* __restrict__ out) {
    extern __shared__ float smem[];
    float* sX = smem;                 // 64  x 76
    float* sU = sX + HH  * US;        // 128 x 76
    float* sT = sU + HH2 * US;        // 128 x 76

    const int tid = threadIdx.x;
    const int img = blockIdx.x;                    // b*256 + c
    const float scale = mask[img & (NCH - 1)];

    const float* __restrict__ xg = x + (size_t)img * (HH * HH);
    float* __restrict__ og = out + (size_t)img * (HH2 * HH2);

#if USE_ASYNC_LDS
    // Async B128 global->LDS: rows are 64 floats = 16 aligned 16B chunks.
    #pragma unroll
    for (int i = 0; i < 4; ++i) {                  // x: 1024 chunks
        int j = tid + i * 256;
        float* dst = sX + (j >> 4) * US + (j & 15) * 4;
        __builtin_amdgcn_global_load_async_to_lds_b128(
            (v4i*)(xg + j * 4), (v4i*)dst, 0, 0);
    }
    #pragma unroll
    for (int i = 0; i < 8; ++i) {                  // U: 2048 chunks
        int j = tid + i * 256;
        float* dst = sU + (j >> 4) * US + (j & 15) * 4;
        __builtin_amdgcn_global_load_async_to_lds_b128(
            (v4i*)(Ug + j * 4), (v4i*)dst, 0, 0);
    }
#if __has_builtin(__builtin_amdgcn_s_wait_asynccnt)
    __builtin_amdgcn_s_wait_asynccnt(0);
#else
    asm volatile("s_wait_asynccnt 0" ::: "memory");
#endif
#else
    #pragma unroll
    for (int i = 0; i < 16; ++i) {
        int e = tid + i * 256;
        sX[(e >> 6) * US + (e & 63)] = xg[e];
    }
    #pragma unroll
    for (int i = 0; i < 32; ++i) {
        int e = tid + i * 256;
        sU[(e >> 6) * US + (e & 63)] = Ug[e];
    }
#endif
    __syncthreads();

    const int w    = tid >> 5;        // wave id 0..7
    const int lane = tid & 31;
    const int half = lane >> 4;       // 0: lanes 0-15, 1: lanes 16-31
    const int lr   = lane & 15;

    // ---- GEMM1: T[16w..16w+15][0..63] = U-strip @ x ----
    // A frag (16x4 f32): lanes 0-15 -> (A[m][k0],A[m][k0+1]); hi lanes k0+2.
    // B frag (4x16 f32): lanes 0-15 -> (B[k0][n],B[k0+1][n]); hi lanes k0+2.
    {
        const float* aU = &sU[(w * 16 + lr) * US + 2 * half];
        v8f acc[4] = {{}, {}, {}, {}};
        #pragma unroll
        for (int k0 = 0; k0 < HH; k0 += 4) {
            v2f a;
            a.x = aU[k0];
            a.y = aU[k0 + 1];
            const float* bX = &sX[(k0 + 2 * half) * US + lr];
            #pragma unroll
            for (int nt = 0; nt < 4; ++nt) {
                v2f b;
                b.x = bX[nt * 16];
                b.y = bX[nt * 16 + US];
                acc[nt] = WMMA_F32(a, b, acc[nt]);
            }
        }
        // D layout: VGPR r -> row (16w + 8*half + r), lane -> col (nt*16 + lr)
        float* t = &sT[(w * 16 + half * 8) * US + lr];
        #pragma unroll
        for (int nt = 0; nt < 4; ++nt)
            #pragma unroll
            for (int r = 0; r < 8; ++r)
                t[r * US + nt * 16] = acc[nt][r];
    }

    // ---- GEMM2: out-strip = T-strip @ U^T, scaled by mask[c] ----
    // B[l][m] = U[m][l]: lane m reads contiguous pairs of U row m.
    {
        const float* aT = &sT[(w * 16 + lr) * US + 2 * half];
        const float* bU = &sU[lr * US + 2 * half];
        v8f acc[8] = {{}, {}, {}, {}, {}, {}, {}, {}};
        #pragma unroll
        for (int k0 = 0; k0 < HH; k0 += 4) {
            v2f a;
            a.x = aT[k0];
            a.y = aT[k0 + 1];
            #pragma unroll
            for (int nt = 0; nt < 8; ++nt) {
                v2f b;
                b.x = bU[nt * 16 * US + k0];
                b.y = bU[nt * 16 * US + k0 + 1];
                acc[nt] = WMMA_F32(a, b, acc[nt]);
            }
        }
        float* o = og + (size_t)(w * 16 + half * 8) * HH2 + lr;
        #pragma unroll
        for (int nt = 0; nt < 8; ++nt)
            #pragma unroll
            for (int r = 0; r < 8; ++r)
                o[(size_t)r * HH2 + nt * 16] = acc[nt][r] * scale;
    }
}

// ---------------------------------------------------------------------------
extern "C" void kernel_launch(void* const* d_in, const int* in_sizes, int n_in,
                              void* d_out, int out_size, void* d_ws, size_t ws_size,
                              hipStream_t stream) {
    const float* x    = (const float*)d_in[0];   // [16,256,64,64] f32
    const float* mask = (const float*)d_in[1];   // [1,256,1,1]   f32
    float* out = (float*)d_out;                  // [16,256,128,128] f32
    float* U   = (float*)d_ws;                   // 128*64 floats scratch

    build_U_kernel<<<(HH2 * HH + 255) / 256, 256, 0, stream>>>(U);

    const int nimg = in_sizes[0] / (HH * HH);    // 16*256 = 4096
    const size_t shmem = (size_t)(HH + HH2 + HH2) * US * sizeof(float); // ~95 KB
    dct_fsr_kernel<<<nimg, 256, shmem, stream>>>(x, mask, U, out);
}